// DeeperGCN_14628658610252
// MI455X (gfx1250) — compile-verified
//
#include <hip/hip_runtime.h>

// DeeperGCN forward for MI455X (gfx1250, wave32).
// fp32 WMMA (V_WMMA_F32_16X16X4_F32) for encoder + per-node MLP GEMMs;
// 2-pass fused edge softmax aggregation (segment-max, then joint sum of a and m*a).

#define NNODES 100000
#define NEDGES 1000000
#define FIN    500
#define H      64
#define H2     128
#define EPS_MSG 1e-7f
#define LN_EPS  1e-5f

typedef __attribute__((ext_vector_type(2))) float v2f;
typedef __attribute__((ext_vector_type(8))) float v8f;

__device__ __forceinline__ v8f wmma_f32(v2f a, v2f b, v8f c) {
  // D = A(16x4,f32) * B(4x16,f32) + C(16x16,f32)
  return __builtin_amdgcn_wmma_f32_16x16x4_f32(false, a, false, b, (short)0, c,
                                               false, false);
}

// ---------------- Encoder: h[N,64] = x[N,500] @ W[500,64] + b ----------------
__global__ __launch_bounds__(256) void enc_gemm_kernel(
    const float* __restrict__ x, const float* __restrict__ W,
    const float* __restrict__ bias, float* __restrict__ h) {
  const int lane = threadIdx.x & 31;
  const int wave = threadIdx.x >> 5;   // 8 waves: 2 row-tiles x 4 col-tiles
  const int m0 = blockIdx.x * 32 + (wave >> 2) * 16;
  const int n0 = (wave & 3) * 16;
  const int lm = lane & 15;
  const int lh = lane >> 4;            // 0/1 -> K sub-pair
  v8f acc = {};
  const float* xrow = x + (size_t)(m0 + lm) * FIN;
  for (int k = 0; k < FIN; k += 4) {
    const int kk = k + 2 * lh;
    v2f a; a.x = xrow[kk];            a.y = xrow[kk + 1];
    v2f b; b.x = W[kk * H + n0 + lm]; b.y = W[(kk + 1) * H + n0 + lm];
    acc = wmma_f32(a, b, acc);
  }
  const float bv = bias[n0 + lm];
#pragma unroll
  for (int r = 0; r < 8; ++r)
    h[(size_t)(m0 + r + 8 * lh) * H + n0 + lm] = acc[r] + bv;
}

// ---------------- Segment buffer init ----------------
__global__ __launch_bounds__(256) void init_seg_kernel(
    float* __restrict__ zmax, float* __restrict__ denom,
    float* __restrict__ numer, int n) {
  const int i = blockIdx.x * blockDim.x + threadIdx.x;
  if (i < n) { zmax[i] = -3.0e38f; denom[i] = 0.f; numer[i] = 0.f; }
}

__device__ __forceinline__ void atomicMaxF(float* addr, float val) {
  int* ia = (int*)addr;
  int old = __float_as_int(*addr);
  while (__int_as_float(old) < val) {
    const int assumed = old;
    old = atomicCAS(ia, assumed, __float_as_int(val));
    if (old == assumed) break;
  }
}

// ---------------- Edge pass 1: zmax[dst] = segmax(m*t) ----------------
__global__ __launch_bounds__(256) void edge_max_kernel(
    const int* __restrict__ ei, const float* __restrict__ h,
    float* __restrict__ zmax, const float* __restrict__ t, int layer) {
  const long long idx = (long long)blockIdx.x * blockDim.x + threadIdx.x;
  if (idx >= (long long)NEDGES * 32) return;
  const int e = (int)(idx >> 5);
  const int c = ((int)idx & 31) * 2;
  const int s = ei[e];
  const int d = ei[NEDGES + e];
  const float tv = t[layer];
  const float2 hv = *(const float2*)(h + (size_t)s * H + c);
  atomicMaxF(zmax + (size_t)d * H + c,     (fmaxf(hv.x, 0.f) + EPS_MSG) * tv);
  atomicMaxF(zmax + (size_t)d * H + c + 1, (fmaxf(hv.y, 0.f) + EPS_MSG) * tv);
}

// ---------------- Edge pass 2: denom = seg_sum(a), numer = seg_sum(m*a) -----
// (agg = numer/denom since denom is segment-constant -> saves a full pass)
__global__ __launch_bounds__(256) void edge_sum_kernel(
    const int* __restrict__ ei, const float* __restrict__ h,
    const float* __restrict__ zmax, float* __restrict__ denom,
    float* __restrict__ numer, const float* __restrict__ t, int layer) {
  const long long idx = (long long)blockIdx.x * blockDim.x + threadIdx.x;
  if (idx >= (long long)NEDGES * 32) return;
  const int e = (int)(idx >> 5);
  const int c = ((int)idx & 31) * 2;
  const int s = ei[e];
  const int d = ei[NEDGES + e];
  const float tv = t[layer];
  const float2 hv = *(const float2*)(h + (size_t)s * H + c);
  const size_t g0 = (size_t)d * H + c;
  const float m0v = fmaxf(hv.x, 0.f) + EPS_MSG;
  const float m1v = fmaxf(hv.y, 0.f) + EPS_MSG;
  const float a0 = __expf(m0v * tv - zmax[g0]);
  const float a1 = __expf(m1v * tv - zmax[g0 + 1]);
  atomicAdd(denom + g0,     a0);
  atomicAdd(numer + g0,     m0v * a0);
  atomicAdd(denom + g0 + 1, a1);
  atomicAdd(numer + g0 + 1, m1v * a1);
}

// ---------------- Fused per-node MLP (+ outer LN/ReLU/residual) -------------
#define SMH 65
#define SMM 130
__global__ __launch_bounds__(256) void mlp_kernel(
    const float* __restrict__ h, const float* __restrict__ numer,
    const float* __restrict__ denom,
    const float* __restrict__ W1, const float* __restrict__ b1,
    const float* __restrict__ g1, const float* __restrict__ be1,
    const float* __restrict__ W2, const float* __restrict__ b2,
    const float* __restrict__ ng, const float* __restrict__ nb,
    float* __restrict__ hout, int residual) {
  __shared__ float sh_hin[32 * SMH];
  __shared__ float sh_mid[32 * SMM];
  __shared__ float sh_y[32 * SMH];
  const int tid = threadIdx.x;
  const int lane = tid & 31, wave = tid >> 5;
  const int lm = lane & 15, lh = lane >> 4;
  const int nodeBase = blockIdx.x * 32;

  // Stage hin = h + agg, agg = numer/denom (0 for isolated nodes).
  for (int i = tid; i < 32 * H; i += 256) {
    const int r = i >> 6, c = i & 63;
    const size_t gi = (size_t)(nodeBase + r) * H + c;
    const float den = denom[gi];
    const float agg = den > 0.f ? numer[gi] / den : 0.f;
    sh_hin[r * SMH + c] = h[gi] + agg;
  }
  __syncthreads();

  // GEMM1: mid[32,128] = hin @ W1 + b1   (each wave: one 16-col strip, 2 row tiles)
  {
    const int n0 = wave * 16;
    for (int rt = 0; rt < 2; ++rt) {
      v8f acc = {};
      const float* arow = sh_hin + (rt * 16 + lm) * SMH;
      for (int k = 0; k < H; k += 4) {
        const int kk = k + 2 * lh;
        v2f a; a.x = arow[kk];                  a.y = arow[kk + 1];
        v2f b; b.x = W1[kk * H2 + n0 + lm];     b.y = W1[(kk + 1) * H2 + n0 + lm];
        acc = wmma_f32(a, b, acc);
      }
      const float bv = b1[n0 + lm];
#pragma unroll
      for (int r = 0; r < 8; ++r)
        sh_mid[(rt * 16 + r + 8 * lh) * SMM + n0 + lm] = acc[r] + bv;
    }
  }
  __syncthreads();

  // LayerNorm + ReLU over 128 features per row.
  if (tid < 32) {
    float* row = sh_mid + tid * SMM;
    float s = 0.f, s2 = 0.f;
    for (int c = 0; c < H2; ++c) { const float v = row[c]; s += v; s2 += v * v; }
    const float mu = s * (1.f / H2);
    const float var = s2 * (1.f / H2) - mu * mu;
    const float rs = rsqrtf(var + LN_EPS);
    for (int c = 0; c < H2; ++c)
      row[c] = fmaxf((row[c] - mu) * rs * g1[c] + be1[c], 0.f);
  }
  __syncthreads();

  // GEMM2: y[32,64] = mid @ W2 + b2   (8 tiles of 16x16, one per wave)
  {
    const int m0 = (wave >> 2) * 16, n0 = (wave & 3) * 16;
    v8f acc = {};
    const float* arow = sh_mid + (m0 + lm) * SMM;
    for (int k = 0; k < H2; k += 4) {
      const int kk = k + 2 * lh;
      v2f a; a.x = arow[kk];               a.y = arow[kk + 1];
      v2f b; b.x = W2[kk * H + n0 + lm];   b.y = W2[(kk + 1) * H + n0 + lm];
      acc = wmma_f32(a, b, acc);
    }
    const float bv = b2[n0 + lm];
#pragma unroll
    for (int r = 0; r < 8; ++r)
      sh_y[(m0 + r + 8 * lh) * SMH + n0 + lm] = acc[r] + bv;
  }
  __syncthreads();

  if (!residual) {
    for (int i = tid; i < 32 * H; i += 256) {
      const int r = i >> 6, c = i & 63;
      hout[(size_t)(nodeBase + r) * H + c] = sh_y[r * SMH + c];
    }
  } else {
    // h_new = h + relu(LN(y, ng, nb))
    if (tid < 32) {
      float* row = sh_y + tid * SMH;
      float s = 0.f, s2 = 0.f;
      for (int c = 0; c < H; ++c) { const float v = row[c]; s += v; s2 += v * v; }
      const float mu = s * (1.f / H);
      const float var = s2 * (1.f / H) - mu * mu;
      const float rs = rsqrtf(var + LN_EPS);
      const size_t gbase = (size_t)(nodeBase + tid) * H;
      for (int c = 0; c < H; ++c) {
        const float v = (row[c] - mu) * rs * ng[c] + nb[c];
        hout[gbase + c] = h[gbase + c] + fmaxf(v, 0.f);
      }
    }
  }
}

// ---------------- Head: out[N,3] = relu(LN(h, ng0, nb0)) @ linW + linb ------
__global__ __launch_bounds__(256) void final_kernel(
    const float* __restrict__ h, const float* __restrict__ ng0,
    const float* __restrict__ nb0, const float* __restrict__ linW,
    const float* __restrict__ linb, float* __restrict__ out) {
  const int node = (blockIdx.x * blockDim.x + threadIdx.x) >> 5;
  const int lane = threadIdx.x & 31;
  if (node >= NNODES) return;
  const float* row = h + (size_t)node * H;
  const int c0 = lane * 2;
  const float v0 = row[c0], v1 = row[c0 + 1];
  float s = v0 + v1, s2 = v0 * v0 + v1 * v1;
  for (int off = 16; off; off >>= 1) {
    s += __shfl_xor(s, off);
    s2 += __shfl_xor(s2, off);
  }
  const float mu = s * (1.f / H);
  const float var = s2 * (1.f / H) - mu * mu;
  const float rs = rsqrtf(var + LN_EPS);
  const float f0 = fmaxf((v0 - mu) * rs * ng0[c0] + nb0[c0], 0.f);
  const float f1 = fmaxf((v1 - mu) * rs * ng0[c0 + 1] + nb0[c0 + 1], 0.f);
#pragma unroll
  for (int c = 0; c < 3; ++c) {
    float p = f0 * linW[c0 * 3 + c] + f1 * linW[(c0 + 1) * 3 + c];
    for (int off = 16; off; off >>= 1) p += __shfl_xor(p, off);
    if (lane == 0) out[(size_t)node * 3 + c] = p + linb[c];
  }
}

extern "C" void kernel_launch(void* const* d_in, const int* in_sizes, int n_in,
                              void* d_out, int out_size, void* d_ws,
                              size_t ws_size, hipStream_t stream) {
  (void)in_sizes; (void)n_in; (void)out_size; (void)ws_size;
  const float* x    = (const float*)d_in[0];
  const int*   ei   = (const int*)d_in[1];
  const float* encW = (const float*)d_in[2];
  const float* encB = (const float*)d_in[3];
  const float* W1   = (const float*)d_in[4];
  const float* b1   = (const float*)d_in[5];
  const float* g1   = (const float*)d_in[6];
  const float* be1  = (const float*)d_in[7];
  const float* W2   = (const float*)d_in[8];
  const float* b2   = (const float*)d_in[9];
  const float* t    = (const float*)d_in[10];
  const float* ng   = (const float*)d_in[11];
  const float* nb   = (const float*)d_in[12];
  const float* linW = (const float*)d_in[13];
  const float* linB = (const float*)d_in[14];
  float* out = (float*)d_out;

  const size_t NH = (size_t)NNODES * H;
  float* ws    = (float*)d_ws;
  float* hA    = ws;
  float* hB    = ws + NH;
  float* zmax  = ws + 2 * NH;
  float* denom = ws + 3 * NH;
  float* numer = ws + 4 * NH;

  enc_gemm_kernel<<<NNODES / 32, 256, 0, stream>>>(x, encW, encB, hA);

  const int layers[4] = {0, 0, 1, 2};  // conv before loop uses layer 0 params
  const int resid[4]  = {0, 1, 1, 1};
  float* hin = hA;
  float* hout = hB;
  const int initBlocks = (int)((NH + 255) / 256);
  const int edgeBlocks = (int)(((long long)NEDGES * 32 + 255) / 256);
  for (int i = 0; i < 4; ++i) {
    const int l = layers[i];
    init_seg_kernel<<<initBlocks, 256, 0, stream>>>(zmax, denom, numer, (int)NH);
    edge_max_kernel<<<edgeBlocks, 256, 0, stream>>>(ei, hin, zmax, t, l);
    edge_sum_kernel<<<edgeBlocks, 256, 0, stream>>>(ei, hin, zmax, denom, numer, t, l);
    mlp_kernel<<<NNODES / 32, 256, 0, stream>>>(
        hin, numer, denom, W1 + (size_t)l * H * H2, b1 + l * H2, g1 + l * H2,
        be1 + l * H2, W2 + (size_t)l * H2 * H, b2 + l * H, ng + l * H,
        nb + l * H, hout, resid[i]);
    float* tmp = hin; hin = hout; hout = tmp;
  }
  final_kernel<<<(NNODES * 32) / 256, 256, 0, stream>>>(hin, ng, nb, linW, linB, out);
}